// WaveletFeatureExtractor_20675972563165
// MI455X (gfx1250) — compile-verified
//
#include <hip/hip_runtime.h>
#include <hip/hip_bf16.h>
#include <stdint.h>

// Problem geometry (fixed by the reference: x is [16, 64, 256, 256] f32)
#define B_   16
#define C_   64
#define H_   256
#define W_   256
#define H2_  128
#define W2_  128

#define TILE_OROWS 8                 // output rows per block
#define TILE_IROWS (2 * TILE_OROWS)  // 16 input rows
#define LDS_FLOATS (TILE_IROWS * W_) // 16 * 256 = 4096 floats = 16 KB
#define THREADS    256               // 8 wave32s

// Native clang vector: required by __builtin_nontemporal_store, and gives
// B128 LDS loads / global stores.
typedef float v4f __attribute__((ext_vector_type(4)));

// Haar butterfly: a=x[2i,2j] b=x[2i,2j+1] c=x[2i+1,2j] d=x[2i+1,2j+1]
// LL=(a+b+c+d)/2  LH=(a+b-c-d)/2  HL=(a-b+c-d)/2  HH=(a-b-c+d)/2
#define HAAR(A, Bv, Cv, Dv, LLo, LHo, HLo, HHo)                  \
    do {                                                         \
        float s0 = (A) + (Bv), s1 = (Cv) + (Dv);                 \
        float d0 = (A) - (Bv), d1 = (Cv) - (Dv);                 \
        (LLo) = (s0 + s1) * 0.5f;                                \
        (LHo) = (s0 - s1) * 0.5f;                                \
        (HLo) = (d0 + d1) * 0.5f;                                \
        (HHo) = (d0 - d1) * 0.5f;                                \
    } while (0)

__global__ __launch_bounds__(THREADS) void haar_dwt2_async_kernel(
    const float* __restrict__ x, float* __restrict__ out)
{
    __shared__ float tile[LDS_FLOATS]; // 16 KB staging tile

    const int      tid     = threadIdx.x;
    const unsigned bid     = blockIdx.x;          // B*C*(H2/8) = 16384 blocks
    const int      tileRow = bid & 15;            // 16 strips per (b,c) plane
    const int      c       = (bid >> 4) & 63;
    const int      b       = bid >> 10;

    // ---------------- Stage 1: async DMA global -> LDS (CDNA5 path) --------
    // The 16-input-row strip spans full W, so it is one contiguous 16 KB
    // region: 1024 x B128 chunks, 4 chunks per lane.
    const size_t in_base  = (((size_t)b * C_ + c) * H_ + (size_t)tileRow * TILE_IROWS) * W_;
    const unsigned lds_base = (unsigned)(size_t)(&tile[0]); // low 32 bits == LDS offset

#pragma unroll
    for (int k = 0; k < 4; ++k) {
        const unsigned chunk = (unsigned)tid + (unsigned)k * THREADS; // 0..1023
        const float*   gptr  = x + in_base + (size_t)chunk * 4;       // 16 B per chunk
        const unsigned laddr = lds_base + chunk * 16u;
        asm volatile("global_load_async_to_lds_b128 %0, %1, off"
                     :
                     : "v"(laddr), "v"(gptr)
                     : "memory");
    }
    // Wait for this wave's async DMA, then sync the workgroup so every wave
    // sees the whole tile.
    asm volatile("s_wait_asynccnt 0" ::: "memory");
    __syncthreads();

    // ---------------- Stage 2: butterfly + non-temporal B128 stores ---------
    const int lh = tid >> 5; // 0..7  : local output row
    const int w4 = tid & 31; // 0..31 : 4 output pixels each (cols 4*w4..4*w4+3)

    const v4f t0 = *(const v4f*)&tile[(2 * lh)     * W_ + 8 * w4];
    const v4f t1 = *(const v4f*)&tile[(2 * lh)     * W_ + 8 * w4 + 4];
    const v4f u0 = *(const v4f*)&tile[(2 * lh + 1) * W_ + 8 * w4];
    const v4f u1 = *(const v4f*)&tile[(2 * lh + 1) * W_ + 8 * w4 + 4];

    v4f vll, vlh, vhl, vhh;
    HAAR(t0.x, t0.y, u0.x, u0.y, vll.x, vlh.x, vhl.x, vhh.x);
    HAAR(t0.z, t0.w, u0.z, u0.w, vll.y, vlh.y, vhl.y, vhh.y);
    HAAR(t1.x, t1.y, u1.x, u1.y, vll.z, vlh.z, vhl.z, vhh.z);
    HAAR(t1.z, t1.w, u1.z, u1.w, vll.w, vlh.w, vhl.w, vhh.w);

    const int h2 = tileRow * TILE_OROWS + lh;

    // Outputs are written once and never re-read: non-temporal stores keep
    // 256 MB of dead lines out of L2 (pure streaming workload, zero reuse).
    // LL: [B, C, H2, W2], first in the flat output
    const size_t ll_idx = (((size_t)b * C_ + c) * H2_ + h2) * W2_ + 4 * w4;
    __builtin_nontemporal_store(vll, (v4f*)&out[ll_idx]);

    // high_freq: stack([LH,HL,HH], axis=1) -> [B, 3, C, H2, W2]
    const size_t NLL   = (size_t)B_ * C_ * H2_ * W2_;        // 16,777,216
    const size_t plane = (size_t)C_ * H2_ * W2_;             // 1,048,576
    const size_t inner = ((size_t)c * H2_ + h2) * W2_ + 4 * w4;
    float* hf = out + NLL;
    __builtin_nontemporal_store(vlh, (v4f*)&hf[((size_t)b * 3 + 0) * plane + inner]);
    __builtin_nontemporal_store(vhl, (v4f*)&hf[((size_t)b * 3 + 1) * plane + inner]);
    __builtin_nontemporal_store(vhh, (v4f*)&hf[((size_t)b * 3 + 2) * plane + inner]);
}

extern "C" void kernel_launch(void* const* d_in, const int* in_sizes, int n_in,
                              void* d_out, int out_size, void* d_ws, size_t ws_size,
                              hipStream_t stream)
{
    (void)in_sizes; (void)n_in; (void)out_size; (void)d_ws; (void)ws_size;

    const float* x   = (const float*)d_in[0];
    float*       out = (float*)d_out;

    const int blocks = B_ * C_ * (H2_ / TILE_OROWS); // 16384
    haar_dwt2_async_kernel<<<blocks, THREADS, 0, stream>>>(x, out);
}